// MambaLayer_59863254172043
// MI455X (gfx1250) — compile-verified
//
#include <hip/hip_runtime.h>
#include <math.h>

typedef __attribute__((ext_vector_type(2))) float v2f;
typedef __attribute__((ext_vector_type(8))) float v8f;
typedef __attribute__((ext_vector_type(4))) unsigned int u32x4;
typedef __attribute__((ext_vector_type(8))) int i32x8;
typedef __attribute__((ext_vector_type(4))) int i32x4;

#define BATCH 2
#define CDIM  128
#define LSEQ  8000
#define NROW  (BATCH * LSEQ)   // 16000
#define DI    256
#define DS    16
#define DTR   8
#define EPSV  1e-5f
#define CH    64               // scan chunk length (8000 % 64 == 0)

__device__ __forceinline__ float sigmoidf_(float x) { return 1.0f / (1.0f + __expf(-x)); }
__device__ __forceinline__ float siluf_(float x)    { return x * sigmoidf_(x); }
__device__ __forceinline__ float softplusf_(float x){ return (x > 20.0f) ? x : log1pf(__expf(x)); }

// ---------------------------------------------------------------------------
// TDM helper: issue a 2D-tile tensor_load_to_lds.
//   tile: tile_d1 rows x tile_d0 elements (fp32), global row stride = stride0
//   elements. Tile assumed fully in-bounds (tensor dims == tile dims).
//   6-arg builtin form (clang-23 / therock-10.0 headers).
// ---------------------------------------------------------------------------
__device__ __forceinline__ void tdm_load_2d(const float* gptr, const float* lptr,
                                            unsigned tile_d0, unsigned tile_d1,
                                            unsigned long long stride0)
{
    const unsigned long long ga = (unsigned long long)(uintptr_t)gptr;
    const unsigned lds_off = (unsigned)(uintptr_t)lptr;   // generic LDS ptr: [31:0] = LDS offset

    u32x4 g0;
    g0.x = 1u;                                            // count=1 (valid), user mode
    g0.y = lds_off;                                       // lds_addr (bytes)
    g0.z = (unsigned)ga;                                  // global_addr[31:0]
    g0.w = ((unsigned)(ga >> 32) & 0x01FFFFFFu) | (2u << 30);  // global_addr[56:32] | type=2

    i32x8 g1;
    g1[0] = (int)(2u << 16);                              // wg_mask=0, data_size=2 (4B)
    g1[1] = (int)((tile_d0 & 0xFFFFu) << 16);             // tensor_dim0[15:0]
    g1[2] = (int)((tile_d0 >> 16) | ((tile_d1 & 0xFFFFu) << 16)); // dim0 hi | dim1 lo
    g1[3] = (int)((tile_d1 >> 16) | (tile_d0 << 16));     // dim1 hi | tile_dim0
    g1[4] = (int)(tile_d1 & 0xFFFFu);                     // tile_dim1, tile_dim2=0
    g1[5] = (int)(unsigned)stride0;                       // tensor_dim0_stride[31:0]
    g1[6] = (int)((unsigned)(stride0 >> 32) & 0xFFFFu);   // stride0[47:32], dim1_stride=0
    g1[7] = 0;

    i32x4 gz4 = {0, 0, 0, 0};                             // groups 2/3 unused (2D tile)
    i32x8 gz8 = {0, 0, 0, 0, 0, 0, 0, 0};
    __builtin_amdgcn_tensor_load_to_lds(g0, g1, gz4, gz4, gz8, 0);
}

// ---------------------------------------------------------------------------
// Kernel 1: fused LayerNorm (over C=128) + in_proj GEMM (16 x 512 per block)
//   xz[r][0:256] = xi, xz[r][256:512] = z,  xz = LN(x_flat) @ in_proj_w^T
//   Uses V_WMMA_F32_16X16X4_F32 (full fp32 precision).
// ---------------------------------------------------------------------------
__global__ __launch_bounds__(256) void k_ln_inproj(
    const float* __restrict__ x, const float* __restrict__ gamma,
    const float* __restrict__ beta, const float* __restrict__ W,
    float* __restrict__ xz)
{
    __shared__ float As[16 * 132];        // 16 rows x 128 cols, pitch 132 (bank-friendly)
    __shared__ float red1[16][16];
    __shared__ float red2[16][16];
    __shared__ float s_mu[16], s_rs[16];

    const int r0 = blockIdx.x * 16;       // tiles never cross batch: 8000 % 16 == 0
    const int b  = r0 / LSEQ;
    const int l0 = r0 % LSEQ;
    const int t  = threadIdx.x;
    const int rloc = t & 15;              // row within tile
    const int cg   = t >> 4;              // column group 0..15 (8 cols each)

    // ---- LayerNorm: coalesced along l (lane-major over rows) ----
    const float* xb = x + (size_t)b * CDIM * LSEQ + (l0 + rloc);
    float xv[8];
    float s = 0.f, s2 = 0.f;
#pragma unroll
    for (int i = 0; i < 8; ++i) {
        int c = cg * 8 + i;
        float v = xb[(size_t)c * LSEQ];
        xv[i] = v; s += v; s2 += v * v;
    }
    red1[cg][rloc] = s; red2[cg][rloc] = s2;
    __syncthreads();
    if (t < 16) {
        float su = 0.f, sq = 0.f;
#pragma unroll
        for (int g = 0; g < 16; ++g) { su += red1[g][t]; sq += red2[g][t]; }
        float mu  = su * (1.0f / CDIM);
        float var = sq * (1.0f / CDIM) - mu * mu;
        s_mu[t] = mu; s_rs[t] = rsqrtf(var + EPSV);
    }
    __syncthreads();
    {
        float mu = s_mu[rloc], rs = s_rs[rloc];
#pragma unroll
        for (int i = 0; i < 8; ++i) {
            int c = cg * 8 + i;
            As[rloc * 132 + c] = (xv[i] - mu) * rs * gamma[c] + beta[c];
        }
    }
    __syncthreads();

    // ---- WMMA GEMM: (16x128) @ (128x512); wave w owns N-tiles [4w..4w+3] ----
    const int wave = t >> 5, lane = t & 31;
    const int lhalf = lane >> 4, l16 = lane & 15;
    v8f acc[4];
#pragma unroll
    for (int j = 0; j < 4; ++j) acc[j] = (v8f){0, 0, 0, 0, 0, 0, 0, 0};
    const int n0w = wave * 64;

    for (int k4 = 0; k4 < 128; k4 += 4) {
        const int kk = k4 + 2 * lhalf;
        v2f a;
        a.x = As[l16 * 132 + kk];
        a.y = As[l16 * 132 + kk + 1];
#pragma unroll
        for (int j = 0; j < 4; ++j) {
            const int n0 = n0w + j * 16;
            const float* wp = W + (size_t)(n0 + l16) * 128 + kk;  // in_proj_w[n][k]
            v2f bf; bf.x = wp[0]; bf.y = wp[1];
            acc[j] = __builtin_amdgcn_wmma_f32_16x16x4_f32(
                false, a, false, bf, (short)0, acc[j], false, false);
        }
    }
#pragma unroll
    for (int j = 0; j < 4; ++j) {
        const int ncol = n0w + j * 16 + l16;
#pragma unroll
        for (int v = 0; v < 8; ++v) {
            const int m = v + 8 * lhalf;
            xz[(size_t)(r0 + m) * 512 + ncol] = acc[j][v];
        }
    }
}

// ---------------------------------------------------------------------------
// Kernel 2: causal depthwise conv (d_conv=4) + SiLU  -> xc[r][d]
// ---------------------------------------------------------------------------
__global__ __launch_bounds__(256) void k_conv(
    const float* __restrict__ xz, const float* __restrict__ cw,
    const float* __restrict__ cb, float* __restrict__ xc)
{
    const int idx = blockIdx.x * 256 + threadIdx.x;
    const int d = idx & 255;
    const int r = idx >> 8;
    if (r >= NROW) return;
    const int l = r % LSEQ;
    float acc = cb[d];
#pragma unroll
    for (int k = 0; k < 4; ++k) {
        const int ll = l - 3 + k;
        if (ll >= 0) acc += cw[d * 4 + k] * xz[(size_t)(r - 3 + k) * 512 + d];
    }
    xc[(size_t)r * 256 + d] = siluf_(acc);
}

// ---------------------------------------------------------------------------
// Kernel 3: dbc = xc @ x_proj_w^T   (16000 x 40, K=256) — small-N dot kernel
// ---------------------------------------------------------------------------
__global__ __launch_bounds__(256) void k_xproj(
    const float* __restrict__ xc, const float* __restrict__ xpw,
    float* __restrict__ dbc)
{
    const int idx = blockIdx.x * 256 + threadIdx.x;
    if (idx >= NROW * 40) return;
    const int n = idx % 40;
    const int r = idx / 40;
    const float4* xr = reinterpret_cast<const float4*>(xc + (size_t)r * 256);
    const float4* wr = reinterpret_cast<const float4*>(xpw + (size_t)n * 256);
    float acc = 0.f;
#pragma unroll 8
    for (int i = 0; i < 64; ++i) {
        float4 a = xr[i], w = wr[i];
        acc += a.x * w.x + a.y * w.y + a.z * w.z + a.w * w.w;
    }
    dbc[(size_t)r * 40 + n] = acc;
}

// ---------------------------------------------------------------------------
// Kernel 4: dt = softplus(dbc[:, :8] @ dt_proj_w^T + dt_proj_b)
// ---------------------------------------------------------------------------
__global__ __launch_bounds__(256) void k_dt(
    const float* __restrict__ dbc, const float* __restrict__ dtw,
    const float* __restrict__ dtbias, float* __restrict__ dtb)
{
    const int idx = blockIdx.x * 256 + threadIdx.x;
    const int d = idx & 255;
    const int r = idx >> 8;
    if (r >= NROW) return;
    float acc = dtbias[d];
#pragma unroll
    for (int rr = 0; rr < DTR; ++rr)
        acc += dbc[(size_t)r * 40 + rr] * dtw[d * DTR + rr];
    dtb[(size_t)r * 256 + d] = softplusf_(acc);
}

// ---------------------------------------------------------------------------
// Kernel 5: selective scan with TDM double-buffered LDS staging.
//   Block = 256 threads = one batch x 16 channels (lane = state index n).
//   Wave 0 issues 4 tensor_load_to_lds per 64-step chunk (dt, xc, z, dbc),
//   waits TENSORcnt, barrier publishes; recurrence runs entirely from LDS.
//   h = exp(dt*A)*h + dt*B*xc ; y = <h,C> via shuffle reduction over 16 lanes
//   fused tail: y = (y + xc*D) * silu(z).
// ---------------------------------------------------------------------------
__global__ __launch_bounds__(256) void k_scan(
    const float* __restrict__ dtb, const float* __restrict__ xc,
    const float* __restrict__ dbc, const float* __restrict__ xz,
    const float* __restrict__ A_log, const float* __restrict__ Dp,
    float* __restrict__ yb)
{
    __shared__ float s_dt[2][CH * 16];
    __shared__ float s_xc[2][CH * 16];
    __shared__ float s_z [2][CH * 16];
    __shared__ float s_bc[2][CH * 40];

    const int t = threadIdx.x;
    const int n = t & 15;
    const int dloc = t >> 4;
    const int b = blockIdx.x >> 4;
    const int d0 = (blockIdx.x & 15) << 4;
    const int d = d0 + dloc;

    const float A  = -__expf(A_log[d * DS + n]);
    const float Dd = Dp[d];
    const int rbase = b * LSEQ;
    const bool issuer = (t >> 5) == 0;    // wave 0 only (TDM is per-wave, EXEC ignored)
    const int NC = LSEQ / CH;

    if (issuer) {
        const int r = rbase;              // chunk 0
        tdm_load_2d(dtb + (size_t)r * 256 + d0,       s_dt[0], 16, CH, 256);
        tdm_load_2d(xc  + (size_t)r * 256 + d0,       s_xc[0], 16, CH, 256);
        tdm_load_2d(xz  + (size_t)r * 512 + 256 + d0, s_z [0], 16, CH, 512);
        tdm_load_2d(dbc + (size_t)r * 40,             s_bc[0], 40, CH, 40);
    }

    float h = 0.f;
    for (int c = 0; c < NC; ++c) {
        const int buf = c & 1;
        if (issuer) __builtin_amdgcn_s_wait_tensorcnt(0);   // chunk c landed in LDS
        __syncthreads();                                    // publish to all waves
        if (issuer && (c + 1 < NC)) {                       // prefetch chunk c+1
            const int r = rbase + (c + 1) * CH;
            const int nb = buf ^ 1;
            tdm_load_2d(dtb + (size_t)r * 256 + d0,       s_dt[nb], 16, CH, 256);
            tdm_load_2d(xc  + (size_t)r * 256 + d0,       s_xc[nb], 16, CH, 256);
            tdm_load_2d(xz  + (size_t)r * 512 + 256 + d0, s_z [nb], 16, CH, 512);
            tdm_load_2d(dbc + (size_t)r * 40,             s_bc[nb], 40, CH, 40);
        }
        const int l0 = c * CH;
        for (int l = 0; l < CH; ++l) {
            const float dt = s_dt[buf][l * 16 + dloc];
            const float xv = s_xc[buf][l * 16 + dloc];
            const float Bn = s_bc[buf][l * 40 + DTR + n];
            const float Cn = s_bc[buf][l * 40 + DTR + DS + n];
            const float dA = __expf(dt * A);
            h = dA * h + dt * Bn * xv;
            float acc = h * Cn;
            acc += __shfl_xor(acc, 1);
            acc += __shfl_xor(acc, 2);
            acc += __shfl_xor(acc, 4);
            acc += __shfl_xor(acc, 8);
            if (n == 0) {
                const float z = s_z[buf][l * 16 + dloc];
                yb[(size_t)(rbase + l0 + l) * 256 + d] = (acc + xv * Dd) * siluf_(z);
            }
        }
        __syncthreads();   // everyone done with buf before chunk c+2 overwrites it
    }
}

// ---------------------------------------------------------------------------
// Kernel 6: out = x + y @ out_proj_w^T, scattered back to (B, C, L) layout.
//   WMMA f32 16x16x4, 16-row tile per block, wave w owns N-tile w (N=128).
// ---------------------------------------------------------------------------
__global__ __launch_bounds__(256) void k_outproj(
    const float* __restrict__ yb, const float* __restrict__ Wo,
    const float* __restrict__ x, float* __restrict__ out)
{
    const int r0 = blockIdx.x * 16;
    const int t = threadIdx.x, wave = t >> 5, lane = t & 31;
    const int lhalf = lane >> 4, l16 = lane & 15;
    const int n0 = wave * 16;

    v8f acc = (v8f){0, 0, 0, 0, 0, 0, 0, 0};
    for (int k4 = 0; k4 < 256; k4 += 4) {
        const int kk = k4 + 2 * lhalf;
        v2f a  = *reinterpret_cast<const v2f*>(yb + (size_t)(r0 + l16) * 256 + kk);
        v2f bf = *reinterpret_cast<const v2f*>(Wo + (size_t)(n0 + l16) * 256 + kk);
        acc = __builtin_amdgcn_wmma_f32_16x16x4_f32(
            false, a, false, bf, (short)0, acc, false, false);
    }
    const int b  = r0 / LSEQ;
    const int l0 = r0 % LSEQ;
    const int c  = n0 + l16;
#pragma unroll
    for (int v = 0; v < 8; ++v) {
        const int m = v + 8 * lhalf;
        const size_t idx = (size_t)b * CDIM * LSEQ + (size_t)c * LSEQ + (l0 + m);
        out[idx] = x[idx] + acc[v];
    }
}

// ---------------------------------------------------------------------------
extern "C" void kernel_launch(void* const* d_in, const int* in_sizes, int n_in,
                              void* d_out, int out_size, void* d_ws, size_t ws_size,
                              hipStream_t stream)
{
    (void)in_sizes; (void)n_in; (void)out_size; (void)ws_size;
    const float* x      = (const float*)d_in[0];
    const float* gamma  = (const float*)d_in[1];
    const float* beta   = (const float*)d_in[2];
    const float* Wi     = (const float*)d_in[3];   // (512,128)
    const float* cw     = (const float*)d_in[4];   // (256,1,4)
    const float* cb     = (const float*)d_in[5];
    const float* xpw    = (const float*)d_in[6];   // (40,256)
    const float* dtw    = (const float*)d_in[7];   // (256,8)
    const float* dtbias = (const float*)d_in[8];
    const float* A_log  = (const float*)d_in[9];   // (256,16)
    const float* Dp     = (const float*)d_in[10];
    const float* Wo     = (const float*)d_in[11];  // (128,256)
    float* out = (float*)d_out;

    float* xz  = (float*)d_ws;                        // 16000*512
    float* xc  = xz  + (size_t)NROW * 512;            // 16000*256
    float* dbc = xc  + (size_t)NROW * 256;            // 16000*40
    float* dtb = dbc + (size_t)NROW * 40;             // 16000*256
    float* yb  = dtb + (size_t)NROW * 256;            // 16000*256

    k_ln_inproj<<<NROW / 16, 256, 0, stream>>>(x, gamma, beta, Wi, xz);
    k_conv<<<NROW, 256, 0, stream>>>(xz, cw, cb, xc);
    k_xproj<<<(NROW * 40 + 255) / 256, 256, 0, stream>>>(xc, xpw, dbc);
    k_dt<<<NROW, 256, 0, stream>>>(dbc, dtw, dtbias, dtb);
    k_scan<<<BATCH * 16, 256, 0, stream>>>(dtb, xc, dbc, xz, A_log, Dp, yb);
    k_outproj<<<NROW / 16, 256, 0, stream>>>(yb, Wo, x, out);
}